// ModelNew_1580547974782
// MI455X (gfx1250) — compile-verified
//
#include <hip/hip_runtime.h>
#include <hip/hip_bf16.h>

// Problem constants (from the reference):
//   x: (64, 64, 128, 128) fp32, weight: (64, 128, 4, 4) fp32, bias: (128,)
//   Hout = Wout = 256  ->  inv_hw = 1/65536
//   out: (64, 128, 1, 1) fp32
#define N_BATCH   64
#define C_IN      64
#define C_OUT     128
#define HW_IN     (128 * 128)          // 16384 elements per (n, cin) row
#define INV_HW    (1.0f / 65536.0f)
#define MULT      0.5f

typedef __attribute__((ext_vector_type(2))) float v2f;
typedef __attribute__((ext_vector_type(8))) float v8f;

// ---------------------------------------------------------------------------
// Kernel 1: sx[n*C_IN + cin] = sum over 16384 spatial elements.
// One block (256 threads = 8 waves) per row; each thread does 16 float4 loads
// (global_load_b128), fully coalesced. This kernel carries ~100% of the
// 256 MiB HBM traffic and is the only perf-critical piece (~11.5 us floor).
// ---------------------------------------------------------------------------
__global__ void reduce_x_kernel(const float* __restrict__ x,
                                float* __restrict__ sx) {
    const int row = blockIdx.x;                       // 0 .. 4095 = n*C_IN+cin
    const float4* p =
        reinterpret_cast<const float4*>(x + (size_t)row * HW_IN);

    float s = 0.0f;
#pragma unroll
    for (int i = 0; i < 16; ++i) {                    // 16 * 256 = 4096 vec4s
        float4 v = p[i * 256 + threadIdx.x];
        s += (v.x + v.y) + (v.z + v.w);
    }

    // wave32 reduction
#pragma unroll
    for (int off = 16; off > 0; off >>= 1)
        s += __shfl_down(s, off, 32);

    __shared__ float lds[8];
    const int lane = threadIdx.x & 31;
    const int wave = threadIdx.x >> 5;
    if (lane == 0) lds[wave] = s;
    __syncthreads();
    if (threadIdx.x == 0) {
        float t = 0.0f;
#pragma unroll
        for (int wv = 0; wv < 8; ++wv) t += lds[wv];
        sx[row] = t;
    }
}

// ---------------------------------------------------------------------------
// Kernel 2: sw[cin*C_OUT + cout] = sum of the 16 kernel taps. Trivial.
// Layout is (Cin, Cout) row-major == K-major for the GEMM's B operand.
// ---------------------------------------------------------------------------
__global__ void reduce_w_kernel(const float* __restrict__ w,
                                float* __restrict__ sw) {
    const int idx = blockIdx.x * blockDim.x + threadIdx.x;  // 0 .. 8191
    const float4* p = reinterpret_cast<const float4*>(w + (size_t)idx * 16);
    float4 a = p[0], b = p[1], c = p[2], d = p[3];
    sw[idx] = ((a.x + a.y) + (a.z + a.w)) + ((b.x + b.y) + (b.z + b.w)) +
              ((c.x + c.y) + (c.z + c.w)) + ((d.x + d.y) + (d.z + d.w));
}

// ---------------------------------------------------------------------------
// Kernel 3: acc(64x128) = sx(64x64) @ sw(64x128); out = (acc*INV_HW+bias)*MULT
// One wave32 per 16x16 output tile (grid 8x4 tiles, blockDim=32 so EXEC is
// all ones as WMMA requires). K=64 -> 16 chained V_WMMA_F32_16X16X4_F32 ops.
// fp32 WMMA keeps the accumulation numerically identical to the reference.
//
// Fragment layouts per CDNA5 ISA 7.12.2:
//   A (16x4 f32, 2 VGPRs): lanes 0-15 -> M=lane, V0=K0,V1=K1;
//                          lanes 16-31 -> M=lane-16, V0=K2,V1=K3.
//   B (4x16 f32, 2 VGPRs): lanes 0-15 -> N=lane, V0=K0,V1=K1;
//                          lanes 16-31 -> N=lane-16, V0=K2,V1=K3.
//   D (16x16 f32, 8 VGPRs): Vv lanes 0-15 -> M=v, lanes 16-31 -> M=v+8; N=lane&15.
// ---------------------------------------------------------------------------
__global__ void gemm_wmma_kernel(const float* __restrict__ sx,
                                 const float* __restrict__ sw,
                                 const float* __restrict__ bias,
                                 float* __restrict__ out) {
    const int lane  = threadIdx.x;        // 0..31
    const int l15   = lane & 15;
    const int half  = lane >> 4;          // 0 or 1
    const int nTile = blockIdx.x * 16;    // 8 tiles over C_OUT
    const int mTile = blockIdx.y * 16;    // 4 tiles over N_BATCH

    v8f c = {};

#if __has_builtin(__builtin_amdgcn_wmma_f32_16x16x4_f32)
#pragma unroll
    for (int kk = 0; kk < C_IN; kk += 4) {
        const int ka = kk + half * 2;     // this half-wave's K pair
        v2f a, b;
        a.x = sx[(mTile + l15) * C_IN + ka];
        a.y = sx[(mTile + l15) * C_IN + ka + 1];
        b.x = sw[(size_t)ka       * C_OUT + (nTile + l15)];
        b.y = sw[(size_t)(ka + 1) * C_OUT + (nTile + l15)];
        // 8 args: (neg_a, A, neg_b, B, c_mod, C, reuse_a, reuse_b)
        c = __builtin_amdgcn_wmma_f32_16x16x4_f32(
                false, a, false, b, (short)0, c, false, false);
    }
#else
    // Fallback (also taken by the host compile pass, where the amdgcn
    // builtin is not declared): plain FMA accumulation into the same
    // per-lane fragment layout, so the epilogue below is shared.
#pragma unroll 4
    for (int k = 0; k < C_IN; ++k) {
        const float bv = sw[(size_t)k * C_OUT + (nTile + l15)];
#pragma unroll
        for (int v = 0; v < 8; ++v) {
            const int m = mTile + v + half * 8;
            c[v] += sx[m * C_IN + k] * bv;
        }
    }
#endif

    const float bi = bias[nTile + l15];
#pragma unroll
    for (int v = 0; v < 8; ++v) {
        const int m = mTile + v + half * 8;
        out[m * C_OUT + (nTile + l15)] = (c[v] * INV_HW + bi) * MULT;
    }
}

// ---------------------------------------------------------------------------
extern "C" void kernel_launch(void* const* d_in, const int* in_sizes, int n_in,
                              void* d_out, int out_size, void* d_ws,
                              size_t ws_size, hipStream_t stream) {
    (void)in_sizes; (void)n_in; (void)out_size; (void)ws_size;

    const float* x    = (const float*)d_in[0];   // 64*64*128*128
    const float* w    = (const float*)d_in[1];   // 64*128*4*4
    const float* bias = (const float*)d_in[2];   // 128

    float* out = (float*)d_out;                  // 64*128
    float* sx  = (float*)d_ws;                   // 4096 floats
    float* sw  = sx + N_BATCH * C_IN;            // 8192 floats (48 KiB total)

    reduce_x_kernel<<<N_BATCH * C_IN, 256, 0, stream>>>(x, sx);
    reduce_w_kernel<<<(C_IN * C_OUT) / 256, 256, 0, stream>>>(w, sw);
    gemm_wmma_kernel<<<dim3(C_OUT / 16, N_BATCH / 16), 32, 0, stream>>>(
        sx, sw, bias, out);
}